// SuperpixelEbli_17815524344016
// MI455X (gfx1250) — compile-verified
//
#include <hip/hip_runtime.h>
#include <math.h>

// ---------------- types / helpers ----------------
typedef __attribute__((ext_vector_type(2))) float v2f;
typedef __attribute__((ext_vector_type(8))) float v8f;

__device__ __forceinline__ v8f wmma4(v2f a, v2f b, v8f c) {
  // D = A(16x4) * B(4x16) + C(16x16), all f32
  return __builtin_amdgcn_wmma_f32_16x16x4_f32(false, a, false, b, (short)0, c,
                                               false, false);
}

__device__ __forceinline__ void atomAddF(float* p, float v) {
  __hip_atomic_fetch_add(p, v, __ATOMIC_RELAXED, __HIP_MEMORY_SCOPE_AGENT);
}

#define GG 512
#define OUTC 10

// ---------------- zero fill ----------------
__global__ void fill0_kernel(float* __restrict__ p, int n) {
  int i = blockIdx.x * blockDim.x + threadIdx.x;
  if (i < n) p[i] = 0.0f;
}

// ---------------- SpMM, 32 features: one wave per edge ----------------
__global__ void spmm32_kernel(const int* __restrict__ rows, const int* __restrict__ cols,
                              const float* __restrict__ vals, const float* __restrict__ x,
                              float* __restrict__ y, int nnz) {
  int tid = blockIdx.x * blockDim.x + threadIdx.x;
  int e = tid >> 5;
  int lane = threadIdx.x & 31;
  if (e >= nnz) return;
  // stream-prefetch edge arrays ahead
  __builtin_prefetch(rows + e + 2048, 0, 0);
  __builtin_prefetch(cols + e + 2048, 0, 0);
  __builtin_prefetch(vals + e + 2048, 0, 0);
  int r = rows[e];
  int c = cols[e];
  float v = vals[e];
  atomAddF(&y[(size_t)r * 32 + lane], v * x[(size_t)c * 32 + lane]);
}

// ---------------- SpMM, 10 features: 3 edges per wave ----------------
__global__ void spmm10_kernel(const int* __restrict__ rows, const int* __restrict__ cols,
                              const float* __restrict__ vals, const float* __restrict__ x,
                              float* __restrict__ y, int nnz) {
  int tid = blockIdx.x * blockDim.x + threadIdx.x;
  int wid = tid >> 5;
  int lane = threadIdx.x & 31;
  int sub = lane / 10;          // 0..3
  int f = lane - sub * 10;      // 0..9
  if (sub >= 3) return;
  int e = wid * 3 + sub;
  if (e >= nnz) return;
  __builtin_prefetch(rows + e + 2048, 0, 0);
  int r = rows[e];
  int c = cols[e];
  float v = vals[e];
  atomAddF(&y[(size_t)r * 10 + f], v * x[(size_t)c * 10 + f]);
}

// ---------------------------------------------------------------------------
// linear: (n x INSTRIDE-row) @ W(10 x WK)^T + b, via f32 WMMA 16x16x4.
// Weights staged in LDS zero-padded to 16 x KPAD: rows n>=10 and cols k>=WK are
// zero, so all A/B operand loads are unconditional (padded-A garbage * 0 = 0).
// Output written to hcat[row*32 + coloff + o] (concat layout), plus optional
// leaky-relu copy at stride 10.
// ---------------------------------------------------------------------------
template <int KPAD, int INSTRIDE, int WK>
__global__ void lin_kernel(const float* __restrict__ in, const float* __restrict__ W,
                           const float* __restrict__ b, float* __restrict__ hcat,
                           int coloff, float* __restrict__ l, int n) {
  __shared__ float wlds[16 * KPAD];
  for (int i = threadIdx.x; i < 16 * KPAD; i += blockDim.x) wlds[i] = 0.0f;
  __syncthreads();
  for (int i = threadIdx.x; i < 10 * WK; i += blockDim.x) {
    int o = i / WK;
    int k = i - o * WK;
    wlds[o * KPAD + k] = W[i];
  }
  __syncthreads();

  int wave = (blockIdx.x * blockDim.x + threadIdx.x) >> 5;
  int lane = threadIdx.x & 31;
  int base = wave * 16;
  if (base < n) {                 // wave-uniform (n % 128 == 0): EXEC all-1s
    int hi = lane >> 4;           // which K half-pair
    int mi = lane & 15;           // row (A) / col (B)
    size_t m = (size_t)(base + mi);
    v8f c = {};
#pragma unroll
    for (int s = 0; s < KPAD / 4; ++s) {
      int k0 = 4 * s + 2 * hi;
      v2f a, bb;
      a.x = in[m * INSTRIDE + k0];
      a.y = in[m * INSTRIDE + k0 + 1];
      bb.x = wlds[mi * KPAD + k0];
      bb.y = wlds[mi * KPAD + k0 + 1];
      c = wmma4(a, bb, c);
    }
    if (mi < 10) {
      float bias = b[mi];
#pragma unroll
      for (int v = 0; v < 8; ++v) {
        size_t row = (size_t)(base + hi * 8 + v);
        float val = c[v] + bias;
        hcat[row * 32 + coloff + mi] = val;
        if (l) l[row * 10 + mi] = val > 0.0f ? val : 0.01f * val;
      }
    }
  }
}

// ---------------------------------------------------------------------------
// concat(h1|h2|h3) @ Wl(10x30)^T + bl, pooled by batch into sacc (atomic).
// A comes from hcat (stride 32, cols 30/31 garbage * zero-B), B staged in LDS.
// ---------------------------------------------------------------------------
__global__ void catpool_kernel(const float* __restrict__ hcat, const float* __restrict__ Wl,
                               const float* __restrict__ bl, const int* __restrict__ batch,
                               float* __restrict__ sacc, int n) {
  __shared__ float wlds[16 * 32];
  for (int i = threadIdx.x; i < 16 * 32; i += blockDim.x) wlds[i] = 0.0f;
  __syncthreads();
  for (int i = threadIdx.x; i < 10 * 30; i += blockDim.x) {
    int o = i / 30;
    int k = i - o * 30;
    wlds[o * 32 + k] = Wl[i];
  }
  __syncthreads();

  int wave = (blockIdx.x * blockDim.x + threadIdx.x) >> 5;
  int lane = threadIdx.x & 31;
  int base = wave * 16;
  if (base < n) {
    int hi = lane >> 4;
    int mi = lane & 15;
    size_t m = (size_t)(base + mi);
    v8f c = {};
#pragma unroll
    for (int s = 0; s < 8; ++s) {
      int k0 = 4 * s + 2 * hi;
      v2f a, bb;
      a.x = hcat[m * 32 + k0];
      a.y = hcat[m * 32 + k0 + 1];
      bb.x = wlds[mi * 32 + k0];
      bb.y = wlds[mi * 32 + k0 + 1];
      c = wmma4(a, bb, c);
    }
    if (mi < 10) {
      float bias = bl[mi];
#pragma unroll
      for (int v = 0; v < 8; ++v) {
        int row = base + hi * 8 + v;
        int g = batch[row];
        atomAddF(&sacc[g * 10 + mi], c[v] + bias);
      }
    }
  }
}

// ---------------- segment counts ----------------
__global__ void count_kernel(const int* __restrict__ batch, float* __restrict__ cnt, int n) {
  int i = blockIdx.x * blockDim.x + threadIdx.x;
  if (i < n) atomAddF(&cnt[batch[i]], 1.0f);
}

// ---------------- mean over groups ----------------
__global__ void mean_kernel(const float* __restrict__ sacc, const float* __restrict__ cnt,
                            float* __restrict__ obr) {
  int i = blockIdx.x * blockDim.x + threadIdx.x;  // 3*G*OUT
  if (i >= 3 * GG * OUTC) return;
  int b = i / (GG * OUTC);
  int g = (i / OUTC) % GG;
  obr[i] = sacc[i] / fmaxf(cnt[b * GG + g], 1.0f);
}

// ---------------- final: concat(3 branches) @ Wc(10x30)^T + bc, softmax ----------------
__global__ void final_kernel(const float* __restrict__ obr, const float* __restrict__ Wc,
                             const float* __restrict__ bc, float* __restrict__ out) {
  int g = blockIdx.x * blockDim.x + threadIdx.x;
  if (g >= GG) return;
  float in[30];
#pragma unroll
  for (int br = 0; br < 3; ++br)
#pragma unroll
    for (int j = 0; j < 10; ++j) in[br * 10 + j] = obr[(br * GG + g) * 10 + j];
  float logit[10];
  float mx = -3.0e38f;
#pragma unroll
  for (int o = 0; o < 10; ++o) {
    float acc = bc[o];
#pragma unroll
    for (int j = 0; j < 30; ++j) acc += Wc[o * 30 + j] * in[j];
    logit[o] = acc;
    mx = fmaxf(mx, acc);
  }
  float sum = 0.0f;
#pragma unroll
  for (int o = 0; o < 10; ++o) {
    logit[o] = __expf(logit[o] - mx);
    sum += logit[o];
  }
  float inv = 1.0f / sum;
#pragma unroll
  for (int o = 0; o < 10; ++o) out[(size_t)g * 10 + o] = logit[o] * inv;
}

// ---------------- host ----------------
static inline int nblk(long long threads, int bs) {
  return (int)((threads + bs - 1) / bs);
}

extern "C" void kernel_launch(void* const* d_in, const int* in_sizes, int n_in,
                              void* d_out, int out_size, void* d_ws, size_t ws_size,
                              hipStream_t stream) {
  (void)in_sizes; (void)n_in; (void)out_size; (void)ws_size;
  const int Ns[3]   = {262144, 524288, 131072};
  const int NNZs[3] = {2097152, 4194304, 1048576};
  const size_t NMAX = 524288;

  float* ws   = (float*)d_ws;
  float* y    = ws;                          // NMAX*32  spmm layer-1 output
  float* hcat = y + NMAX * 32;               // NMAX*32  concat(h1|h2|h3) + 2 pad cols
  float* lbuf = hcat + NMAX * 32;            // NMAX*10  leaky buffer
  float* yv   = lbuf + NMAX * 10;            // NMAX*10  spmm layer-2/3 output
  float* sacc = yv + NMAX * 10;              // 3*G*OUT  pooled sums
  float* cnt  = sacc + 3 * GG * OUTC;        // 3*G      group counts
  float* obr  = cnt + 3 * GG;                // 3*G*OUT  pooled means

  // zero pooled accumulators + counts (contiguous)
  fill0_kernel<<<nblk(3 * GG * OUTC + 3 * GG, 256), 256, 0, stream>>>(
      sacc, 3 * GG * OUTC + 3 * GG);

  for (int br = 0; br < 3; ++br) {
    const int*   rows  = (const int*)d_in[br * 5 + 0];
    const int*   cols  = (const int*)d_in[br * 5 + 1];
    const float* vals  = (const float*)d_in[br * 5 + 2];
    const float* X     = (const float*)d_in[br * 5 + 3];
    const int*   batch = (const int*)d_in[br * 5 + 4];
    const float* W1 = (const float*)d_in[15 + br * 8 + 0];
    const float* b1 = (const float*)d_in[15 + br * 8 + 1];
    const float* W2 = (const float*)d_in[15 + br * 8 + 2];
    const float* b2 = (const float*)d_in[15 + br * 8 + 3];
    const float* W3 = (const float*)d_in[15 + br * 8 + 4];
    const float* b3 = (const float*)d_in[15 + br * 8 + 5];
    const float* Wl = (const float*)d_in[15 + br * 8 + 6];
    const float* bl = (const float*)d_in[15 + br * 8 + 7];
    int n = Ns[br], nnz = NNZs[br];

    // layer 1: y = L @ X (F=32); hcat[:,0:10] = y@W1^T+b1 ; lbuf = leaky
    fill0_kernel<<<nblk((long long)n * 32, 256), 256, 0, stream>>>(y, n * 32);
    spmm32_kernel<<<nblk((long long)nnz * 32, 256), 256, 0, stream>>>(
        rows, cols, vals, X, y, nnz);
    lin_kernel<32, 32, 32><<<nblk((long long)n * 2, 256), 256, 0, stream>>>(
        y, W1, b1, hcat, 0, lbuf, n);

    // layer 2: yv = L @ leaky(h1); hcat[:,10:20] = yv@W2^T+b2 ; lbuf = leaky
    fill0_kernel<<<nblk((long long)n * 10, 256), 256, 0, stream>>>(yv, n * 10);
    spmm10_kernel<<<nblk((long long)((nnz + 2) / 3) * 32, 256), 256, 0, stream>>>(
        rows, cols, vals, lbuf, yv, nnz);
    lin_kernel<12, 10, 10><<<nblk((long long)n * 2, 256), 256, 0, stream>>>(
        yv, W2, b2, hcat, 10, lbuf, n);

    // layer 3: yv = L @ leaky(h2); hcat[:,20:30] = yv@W3^T+b3
    fill0_kernel<<<nblk((long long)n * 10, 256), 256, 0, stream>>>(yv, n * 10);
    spmm10_kernel<<<nblk((long long)((nnz + 2) / 3) * 32, 256), 256, 0, stream>>>(
        rows, cols, vals, lbuf, yv, nnz);
    lin_kernel<12, 10, 10><<<nblk((long long)n * 2, 256), 256, 0, stream>>>(
        yv, W3, b3, hcat, 20, (float*)nullptr, n);

    // concat @ Wl^T + bl, pooled into sacc[br]; plus counts
    catpool_kernel<<<nblk((long long)n * 2, 256), 256, 0, stream>>>(
        hcat, Wl, bl, batch, sacc + br * GG * OUTC, n);
    count_kernel<<<nblk(n, 256), 256, 0, stream>>>(batch, cnt + br * GG, n);
  }

  const float* Wc = (const float*)d_in[39];
  const float* bc = (const float*)d_in[40];

  mean_kernel<<<nblk(3 * GG * OUTC, 256), 256, 0, stream>>>(sacc, cnt, obr);
  final_kernel<<<nblk(GG, 256), 256, 0, stream>>>(obr, Wc, bc, (float*)d_out);
}